// MoELoRALayer_79937931313781
// MI455X (gfx1250) — compile-verified
//
#include <hip/hip_runtime.h>
#include <math.h>

// ---------------------------------------------------------------------------
// MoE-LoRA: out = x @ Wb^T + bias + sum_e w[t,e]*SCALING * (x @ A_e^T) @ B_e^T
// Strategy: fold LoRA into a K-extension of the base GEMM.
//   hw[t, e*16+r]  = (x @ Aflat^T)[t, e*16+r] * w[t,e] * SCALING
//   Bcat[o, e*16+r]= Bm[e, o, r]
//   out = x @ Wb^T  +  hw @ Bcat^T  + bias
// GEMMs: V_WMMA_F32_16X16X4_F32 (exact fp32, CDNA5 matrix pipe) with
// double-buffered LDS fed by GLOBAL_LOAD_ASYNC_TO_LDS_B128 (ASYNCcnt).
// ---------------------------------------------------------------------------

typedef float v8f __attribute__((ext_vector_type(8)));
typedef float v2f __attribute__((ext_vector_type(2)));
typedef int   v4i __attribute__((vector_size(16)));   // matches builtin param

typedef __attribute__((address_space(1))) v4i* gbl_v4i_p;
typedef __attribute__((address_space(3))) v4i* lds_v4i_p;

static constexpr int   T_TOK = 8192;         // 4 * 2048 tokens
static constexpr int   DIN   = 4096;
static constexpr int   DOUT  = 4096;
static constexpr int   NE    = 8;            // experts
static constexpr int   RK    = 16;           // lora rank
static constexpr int   KL    = NE * RK;      // 128 = lora K extension
static constexpr float SCAL  = 2.0f;         // lora_alpha / r = 32/16

// GEMM tiling: 256 threads = 8 waves (4 in M x 2 in N); wave tile 32x32.
static constexpr int BM  = 128;
static constexpr int BN  = 64;
static constexpr int BK  = 16;
static constexpr int LDK = 20;               // padded LDS k-stride (dwords)

// --------------------------- async global->LDS -----------------------------
#if __has_builtin(__builtin_amdgcn_global_load_async_to_lds_b128) && \
    __has_builtin(__builtin_amdgcn_s_wait_asynccnt)
#define HAS_ASYNC_LDS 1
#else
#define HAS_ASYNC_LDS 0
#endif

__device__ __forceinline__ void copy_b128(const float* __restrict__ gsrc,
                                          float* lds_dst) {
#if HAS_ASYNC_LDS
    __builtin_amdgcn_global_load_async_to_lds_b128(
        (gbl_v4i_p)gsrc, (lds_v4i_p)lds_dst, 0, 0);
#else
    *(float4*)lds_dst = *(const float4*)gsrc;
#endif
}

#if HAS_ASYNC_LDS
#define WAIT_ASYNC(n) __builtin_amdgcn_s_wait_asynccnt(n)
#else
#define WAIT_ASYNC(n) ((void)0)
#endif

// ---------------------------------------------------------------------------
// Stage one (BM x BK) A tile + (BN x BK) B tile into LDS.
// 3 b128 copies per thread-slot => asynccnt += 3 per wave per tile.
// ---------------------------------------------------------------------------
__device__ __forceinline__ void stage_tile(const float* __restrict__ Ag, int lda,
                                           const float* __restrict__ Bg, int ldb,
                                           int mBase, int nBase, int k0,
                                           float* As, float* Bs) {
    const int tid = threadIdx.x;
    #pragma unroll
    for (int i = 0; i < 2; ++i) {
        const int idx = tid + i * 256;          // 0..511
        const int row = idx >> 2;               // 0..127
        const int c4  = idx & 3;                // 0..3 (x4 floats)
        copy_b128(Ag + (size_t)(mBase + row) * lda + k0 + c4 * 4,
                  As + row * LDK + c4 * 4);
    }
    {
        const int row = tid >> 2;               // 0..63
        const int c4  = tid & 3;
        copy_b128(Bg + (size_t)(nBase + row) * ldb + k0 + c4 * 4,
                  Bs + row * LDK + c4 * 4);
    }
}

// ---------------------------------------------------------------------------
// Double-buffered inner loop: acc[2][2] (+)= Atile x Btile^T over K.
// Ag rows indexed by (mBase+row); Bg rows by (nBase+row) (B stored transposed,
// like Wb [DOUT, DIN]).
// ---------------------------------------------------------------------------
__device__ __forceinline__ void gemm_loop(const float* __restrict__ Ag, int lda,
                                          const float* __restrict__ Bg, int ldb,
                                          int K, int mBase, int nBase,
                                          float (&As)[2][BM * LDK],
                                          float (&Bs)[2][BN * LDK],
                                          v8f (&acc)[2][2]) {
    const int tid  = threadIdx.x;
    const int wave = tid >> 5;
    const int lane = tid & 31;
    const int wm   = wave & 3;          // 0..3 -> M offset wm*32
    const int wn   = wave >> 2;         // 0..1 -> N offset wn*32
    const int half = lane >> 4;         // K-half selector for A/B fragments
    const int l16  = lane & 15;

    const int ntiles = K / BK;
    stage_tile(Ag, lda, Bg, ldb, mBase, nBase, 0, As[0], Bs[0]);

    for (int t = 0; t < ntiles; ++t) {
        const int cur = t & 1;
        if (t + 1 < ntiles) {
            stage_tile(Ag, lda, Bg, ldb, mBase, nBase, (t + 1) * BK,
                       As[cur ^ 1], Bs[cur ^ 1]);
            WAIT_ASYNC(3);              // tile t complete; tile t+1 in flight
        } else {
            WAIT_ASYNC(0);
        }
        __syncthreads();

        const float* Ac = As[cur];
        const float* Bc = Bs[cur];
        #pragma unroll
        for (int kb = 0; kb < BK; kb += 4) {
            v2f afr[2], bfr[2];
            #pragma unroll
            for (int mi = 0; mi < 2; ++mi)
                afr[mi] = *(const v2f*)(Ac + (wm * 32 + mi * 16 + l16) * LDK
                                           + kb + 2 * half);
            #pragma unroll
            for (int ni = 0; ni < 2; ++ni)
                bfr[ni] = *(const v2f*)(Bc + (wn * 32 + ni * 16 + l16) * LDK
                                           + kb + 2 * half);
            #pragma unroll
            for (int mi = 0; mi < 2; ++mi)
                #pragma unroll
                for (int ni = 0; ni < 2; ++ni)
                    acc[mi][ni] = __builtin_amdgcn_wmma_f32_16x16x4_f32(
                        false, afr[mi], false, bfr[ni],
                        (short)0, acc[mi][ni], false, false);
        }
        __syncthreads();
    }
}

// ---------------------------------------------------------------------------
// Router: logits -> softmax -> top-2 -> renormalize -> dense w[T, 8] * SCAL
// One wave32 per token.
// ---------------------------------------------------------------------------
__global__ __launch_bounds__(256) void moe_router(const float* __restrict__ x,
                                                  const float* __restrict__ Wr,
                                                  float* __restrict__ w) {
    const int wave = threadIdx.x >> 5;
    const int lane = threadIdx.x & 31;
    const int t    = blockIdx.x * 8 + wave;
    const float* xr = x + (size_t)t * DIN;

    float acc[NE];
    #pragma unroll
    for (int e = 0; e < NE; ++e) acc[e] = 0.f;
    for (int k = lane; k < DIN; k += 32) {
        const float xv = xr[k];
        #pragma unroll
        for (int e = 0; e < NE; ++e) acc[e] += xv * Wr[e * DIN + k];
    }
    #pragma unroll
    for (int off = 16; off > 0; off >>= 1) {
        #pragma unroll
        for (int e = 0; e < NE; ++e) acc[e] += __shfl_xor(acc[e], off, 32);
    }
    // softmax (replicated on all lanes)
    float mx = acc[0];
    #pragma unroll
    for (int e = 1; e < NE; ++e) mx = fmaxf(mx, acc[e]);
    float p[NE];
    #pragma unroll
    for (int e = 0; e < NE; ++e) p[e] = __expf(acc[e] - mx);
    // top-2 (ties -> lower index, matches lax.top_k)
    int i0 = 0;
    #pragma unroll
    for (int e = 1; e < NE; ++e) if (p[e] > p[i0]) i0 = e;
    int i1 = (i0 == 0) ? 1 : 0;
    #pragma unroll
    for (int e = 0; e < NE; ++e) if (e != i0 && p[e] > p[i1]) i1 = e;
    const float s2 = p[i0] + p[i1];
    if (lane < NE) {
        const float wv = (lane == i0) ? p[i0] / s2
                       : (lane == i1) ? p[i1] / s2 : 0.f;
        w[(size_t)t * NE + lane] = wv * SCAL;
    }
}

// ---------------------------------------------------------------------------
// Bcat[o, e*16+r] = Bm[e, o, r]   (row-major [DOUT, 128])
// ---------------------------------------------------------------------------
__global__ __launch_bounds__(256) void build_bcat(const float* __restrict__ Bm,
                                                  float* __restrict__ BT) {
    const int idx = blockIdx.x * 256 + threadIdx.x;   // DOUT*KL total
    const int o = idx >> 7;
    const int j = idx & 127;
    const int e = j >> 4;
    const int r = j & 15;
    BT[idx] = Bm[((size_t)e * DOUT + o) * RK + r];
}

// ---------------------------------------------------------------------------
// hw = (x @ Aflat^T) scaled by router weights.  Grid: (T/BM, KL/BN) = (64, 2)
// ---------------------------------------------------------------------------
__global__ __launch_bounds__(256) void moe_h(const float* __restrict__ x,
                                             const float* __restrict__ A,
                                             const float* __restrict__ w,
                                             float* __restrict__ hw) {
    __shared__ __align__(16) float As[2][BM * LDK];
    __shared__ __align__(16) float Bs[2][BN * LDK];
    const int mBase = blockIdx.x * BM;
    const int nBase = blockIdx.y * BN;
    v8f acc[2][2] = {{{0,0,0,0,0,0,0,0},{0,0,0,0,0,0,0,0}},
                     {{0,0,0,0,0,0,0,0},{0,0,0,0,0,0,0,0}}};

    gemm_loop(x, DIN, A, DIN, DIN, mBase, nBase, As, Bs, acc);

    const int wave = threadIdx.x >> 5;
    const int lane = threadIdx.x & 31;
    const int wm = wave & 3, wn = wave >> 2;
    const int half = lane >> 4, l16 = lane & 15;
    #pragma unroll
    for (int mi = 0; mi < 2; ++mi) {
        #pragma unroll
        for (int ni = 0; ni < 2; ++ni) {
            const int j = nBase + wn * 32 + ni * 16 + l16;   // 0..127
            const int e = j >> 4;
            const int mrow = mBase + wm * 32 + mi * 16 + half * 8;
            #pragma unroll
            for (int v = 0; v < 8; ++v) {
                const int t = mrow + v;
                hw[(size_t)t * KL + j] = acc[mi][ni][v] * w[(size_t)t * NE + e];
            }
        }
    }
}

// ---------------------------------------------------------------------------
// Main: out = x @ Wb^T  +  hw @ Bcat^T  + bias.  Grid: (64, 64)
// ---------------------------------------------------------------------------
__global__ __launch_bounds__(256) void moe_main(const float* __restrict__ x,
                                                const float* __restrict__ Wb,
                                                const float* __restrict__ bias,
                                                const float* __restrict__ hw,
                                                const float* __restrict__ BT,
                                                float* __restrict__ out) {
    __shared__ __align__(16) float As[2][BM * LDK];
    __shared__ __align__(16) float Bs[2][BN * LDK];
    const int mBase = blockIdx.x * BM;
    const int nBase = blockIdx.y * BN;
    v8f acc[2][2] = {{{0,0,0,0,0,0,0,0},{0,0,0,0,0,0,0,0}},
                     {{0,0,0,0,0,0,0,0},{0,0,0,0,0,0,0,0}}};

    gemm_loop(x,  DIN, Wb, DIN, DIN, mBase, nBase, As, Bs, acc);   // base GEMM
    gemm_loop(hw, KL,  BT, KL,  KL,  mBase, nBase, As, Bs, acc);   // lora K=128

    const int wave = threadIdx.x >> 5;
    const int lane = threadIdx.x & 31;
    const int wm = wave & 3, wn = wave >> 2;
    const int half = lane >> 4, l16 = lane & 15;
    #pragma unroll
    for (int mi = 0; mi < 2; ++mi) {
        #pragma unroll
        for (int ni = 0; ni < 2; ++ni) {
            const int n = nBase + wn * 32 + ni * 16 + l16;
            const int mrow = mBase + wm * 32 + mi * 16 + half * 8;
            const float bv = bias[n];
            #pragma unroll
            for (int v = 0; v < 8; ++v)
                out[(size_t)(mrow + v) * DOUT + n] = acc[mi][ni][v] + bv;
        }
    }
}

// ---------------------------------------------------------------------------
extern "C" void kernel_launch(void* const* d_in, const int* in_sizes, int n_in,
                              void* d_out, int out_size, void* d_ws, size_t ws_size,
                              hipStream_t stream) {
    const float* x    = (const float*)d_in[0];   // [4,2048,4096]
    const float* Wb   = (const float*)d_in[1];   // [4096,4096]
    const float* bias = (const float*)d_in[2];   // [4096]
    const float* Wr   = (const float*)d_in[3];   // [8,4096]
    const float* A    = (const float*)d_in[4];   // [8,16,4096] == [128,4096]
    const float* Bm   = (const float*)d_in[5];   // [8,4096,16]
    float* out = (float*)d_out;                  // [8192,4096]

    // Workspace carve-up (all float32)
    float* w  = (float*)d_ws;                       // T_TOK * 8      (256 KB)
    float* hw = w  + (size_t)T_TOK * NE;            // T_TOK * 128    (4 MB)
    float* BT = hw + (size_t)T_TOK * KL;            // DOUT * 128     (2 MB)

    // 1) router weights (dense, renormalized, pre-scaled)
    moe_router<<<T_TOK / 8, 256, 0, stream>>>(x, Wr, w);
    // 2) Bcat transpose
    build_bcat<<<(DOUT * KL) / 256, 256, 0, stream>>>(Bm, BT);
    // 3) hw = (x @ Aflat^T) * w          grid (64, 2)
    moe_h<<<dim3(T_TOK / BM, KL / BN), 256, 0, stream>>>(x, A, w, hw);
    // 4) out = x@Wb^T + hw@Bcat^T + bias grid (64, 64)
    moe_main<<<dim3(T_TOK / BM, DOUT / BN), 256, 0, stream>>>(x, Wb, bias, hw, BT, out);
}